// SSDBoxHead_74517682585727
// MI455X (gfx1250) — compile-verified
//
#include <hip/hip_runtime.h>

// ---------------- constants (match reference) ----------------
#define B_     32
#define C_     81
#define FG     80          // foreground classes
#define NBINS  1024
#define CAP    4096        // candidate cap per batch before sort
#define PRE    400         // PRE_NMS_K
#define MAXD   100         // MAX_DET
#define CONF   0.01f
#define NMSTHR 0.45f
#define OFFMUL 1281.0f     // max(H,W)+1
#define WIMG   1280.0f
#define HIMG   1024.0f

typedef float v2f __attribute__((ext_vector_type(2)));
typedef float v8f __attribute__((ext_vector_type(8)));

// ---------------- init: zero histograms + counters ----------------
__global__ void k_init(unsigned* __restrict__ hist, int* __restrict__ cnt) {
    int t = blockIdx.x * blockDim.x + threadIdx.x;
    if (t < B_ * NBINS) hist[t] = 0u;
    if (t < B_) cnt[t] = 0;
}

// ---------------- decode boxes to pixel-scaled corners ----------------
__global__ void k_decode(const float* __restrict__ loc, const float* __restrict__ pri,
                         float* __restrict__ dec, int P) {
    int t = blockIdx.x * blockDim.x + threadIdx.x;
    if (t >= B_ * P) return;
    int p = t % P;
    const float* l  = loc + (size_t)t * 4;
    const float* pr = pri + (size_t)p * 4;
    float px = pr[0], py = pr[1], pw = pr[2], ph = pr[3];
    float cx = l[0] * 0.1f * pw + px;
    float cy = l[1] * 0.1f * ph + py;
    float w  = __expf(l[2] * 0.2f) * pw;
    float h  = __expf(l[3] * 0.2f) * ph;
    float* o = dec + (size_t)t * 4;
    o[0] = (cx - 0.5f * w) * WIMG;
    o[1] = (cy - 0.5f * h) * HIMG;
    o[2] = (cx + 0.5f * w) * WIMG;
    o[3] = (cy + 0.5f * h) * HIMG;
}

// ---------------- fused softmax + (MODE0: histogram | MODE1: compact) ----------------
// Each wave stages one prior's 81 logits into LDS via global_load_async_to_lds_b128
// (lanes 0..19 each move 16B = floats 0..79; class 80 read directly). wave32 shuffles
// do the max/sum reductions.
template <int MODE>
__global__ void __launch_bounds__(256)
k_scores(const float* __restrict__ logits, unsigned* __restrict__ hist,
         const int* __restrict__ thr, int* __restrict__ cnt,
         float* __restrict__ candS, int* __restrict__ candI, int P) {
    __shared__ float    stage[8 * 128];   // 8 waves * 512B staging
    __shared__ unsigned lhist[NBINS];
    const int b    = blockIdx.y;
    const int wave = threadIdx.x >> 5;
    const int lane = threadIdx.x & 31;

    if (MODE == 0) {
        for (int i = threadIdx.x; i < NBINS; i += blockDim.x) lhist[i] = 0u;
        __syncthreads();
    }
    const int tb = (MODE == 1) ? thr[b] : 0;

    float* slot = &stage[wave * 128];
    const unsigned ldsBase = (unsigned)(unsigned long long)(void*)slot; // low 32b = LDS addr
    const int pbase = blockIdx.x * 64 + wave * 8;

    for (int it = 0; it < 8; ++it) {
        const int p = pbase + it;
        if (p >= P) break;
        const float* gp = logits + ((size_t)b * P + p) * C_;
        unsigned long long ga = (unsigned long long)gp + ((lane < 20) ? (unsigned)(lane * 16) : 0u);
        unsigned dst = ldsBase + lane * 16;
        // make sure prior ds reads of this slot retired, then async-stage 512B, then wait
        asm volatile("s_wait_dscnt 0" ::: "memory");
        asm volatile("global_load_async_to_lds_b128 %0, %1, off" :: "v"(dst), "v"(ga) : "memory");
        asm volatile("s_wait_asynccnt 0" ::: "memory");

        float v0 = slot[lane];                          // class lane      (0..31)
        float v1 = slot[lane + 32];                     // class lane+32   (32..63)
        float v2 = (lane < 16) ? slot[lane + 64]        // class lane+64   (64..79)
                 : ((lane == 16) ? gp[80] : -1e30f);    // class 80 via global
        float m = fmaxf(v0, fmaxf(v1, v2));
        for (int s = 16; s; s >>= 1) m = fmaxf(m, __shfl_xor(m, s, 32));
        float e0 = __expf(v0 - m);
        float e1 = __expf(v1 - m);
        float e2 = (lane <= 16) ? __expf(v2 - m) : 0.0f;
        float sum = e0 + e1 + e2;
        for (int s = 16; s; s >>= 1) sum += __shfl_xor(sum, s, 32);
        const float inv = 1.0f / sum;

        auto doScore = [&](int cidx, float sc) {
            if (cidx >= 1) {
                int bin = (int)(sc * (float)NBINS);
                bin = bin < 0 ? 0 : (bin > NBINS - 1 ? NBINS - 1 : bin);
                if (MODE == 0) {
                    atomicAdd(&lhist[bin], 1u);
                } else if (bin >= tb) {
                    int pos = atomicAdd(&cnt[b], 1);
                    if (pos < CAP) {
                        candS[(size_t)b * CAP + pos] = sc;
                        candI[(size_t)b * CAP + pos] = p * FG + (cidx - 1);
                    }
                }
            }
        };
        doScore(lane,        e0 * inv);
        doScore(lane + 32,   e1 * inv);
        if (lane <= 16) doScore((lane == 16) ? 80 : lane + 64, e2 * inv);
    }

    if (MODE == 0) {
        __syncthreads();
        for (int i = threadIdx.x; i < NBINS; i += blockDim.x) {
            unsigned v = lhist[i];
            if (v) atomicAdd(&hist[(size_t)b * NBINS + i], v);
        }
    }
}

// ---------------- find per-batch threshold bin (suffix count >= 400) ----------------
__global__ void k_thr(const unsigned* __restrict__ hist, int* __restrict__ thr) {
    int b = blockIdx.x;
    if (threadIdx.x == 0) {
        unsigned cum = 0; int t = 0;
        for (int i = NBINS - 1; i >= 0; --i) {
            cum += hist[(size_t)b * NBINS + i];
            if (cum >= PRE) { t = i; break; }
        }
        thr[b] = t;
    }
}

// ---------------- bitonic sort candidates, take top-400, gather boxes ----------------
__global__ void __launch_bounds__(1024)
k_sel(const int* __restrict__ cnt, const float* __restrict__ candS, const int* __restrict__ candI,
      const float* __restrict__ dec, float* __restrict__ selBox, float* __restrict__ selOb,
      int* __restrict__ selCls, float* __restrict__ selScore, float* __restrict__ selArea, int P) {
    __shared__ float sS[CAP];
    __shared__ int   sI[CAP];
    const int b = blockIdx.x;
    const int n = min(cnt[b], CAP);
    for (int s = threadIdx.x; s < CAP; s += blockDim.x) {
        bool v = s < n;
        sS[s] = v ? candS[(size_t)b * CAP + s] : -1.0f;
        sI[s] = v ? candI[(size_t)b * CAP + s] : 0x7FFFFFFF;
    }
    __syncthreads();
    for (int k = 2; k <= CAP; k <<= 1) {
        for (int j = k >> 1; j > 0; j >>= 1) {
            for (int s = threadIdx.x; s < CAP; s += blockDim.x) {
                int ixj = s ^ j;
                if (ixj > s) {
                    float sa = sS[s], sb2 = sS[ixj];
                    int   ia = sI[s], ib2 = sI[ixj];
                    bool less = (sa < sb2) || (sa == sb2 && ia > ib2);   // a after b in desc order
                    bool up   = ((s & k) == 0);
                    if (less == up) { sS[s] = sb2; sS[ixj] = sa; sI[s] = ib2; sI[ixj] = ia; }
                }
            }
            __syncthreads();
        }
    }
    if (threadIdx.x < PRE) {
        int r = threadIdx.x;
        float sc = sS[r];
        int   idx = sI[r];
        bool  valid = (idx != 0x7FFFFFFF);
        int   id2 = valid ? idx : 0;
        int   pidx = id2 / FG;
        int   cls  = id2 % FG + 1;
        const float* cb = dec + ((size_t)b * P + pidx) * 4;
        float x0 = cb[0], y0 = cb[1], x1 = cb[2], y1 = cb[3];
        float off = OFFMUL * (float)cls;
        size_t o4 = ((size_t)b * PRE + r) * 4;
        selBox[o4 + 0] = x0; selBox[o4 + 1] = y0; selBox[o4 + 2] = x1; selBox[o4 + 3] = y1;
        selOb[o4 + 0] = x0 + off; selOb[o4 + 1] = y0 + off;
        selOb[o4 + 2] = x1 + off; selOb[o4 + 3] = y1 + off;
        selCls[b * PRE + r]   = cls;
        selScore[b * PRE + r] = valid ? sc : -1.0f;
        selArea[b * PRE + r]  = (x1 - x0) * (y1 - y0);
    }
}

// ---------------- WMMA: union base matrix U[i][j] = area[i] + area[j] ----------------
// Outer sum as rank-2 GEMM on v_wmma_f32_16x16x4_f32: A row m = {area[m], 1, 0, 0},
// B col n = {1, area[n], 0, 0}^T  =>  D[m][n] = area[m] + area[n].
__global__ void __launch_bounds__(32)
k_umat(const float* __restrict__ selArea, float* __restrict__ U) {
    const int b  = blockIdx.z;
    const int i0 = blockIdx.x * 16;
    const int j0 = blockIdx.y * 16;
    const int lane = threadIdx.x;
    const int nn = lane & 15;
    float rowA = selArea[b * PRE + i0 + nn];
    float colA = selArea[b * PRE + j0 + nn];
    v2f a, bm;
    // A (16x4): lanes 0-15 hold K=0,1 ; lanes 16-31 hold K=2,3 (zeros)
    a.x = (lane < 16) ? rowA : 0.0f;
    a.y = (lane < 16) ? 1.0f : 0.0f;
    // B (4x16): VGPR0 = rows K0 (lanes 0-15, ones) / K1 (lanes 16-31, areas); VGPR1 = K2/K3 zeros
    bm.x = (lane < 16) ? 1.0f : colA;
    bm.y = 0.0f;
    v8f c = {};
    c = __builtin_amdgcn_wmma_f32_16x16x4_f32(false, a, false, bm, (short)0, c, false, false);
    const int mBase = (lane < 16) ? 0 : 8;
#pragma unroll
    for (int r = 0; r < 8; ++r)
        U[((size_t)b * PRE + (i0 + mBase + r)) * PRE + (j0 + nn)] = c[r];
}

// ---------------- NMS scan + final top-100 ----------------
__global__ void __launch_bounds__(512)
k_nms(const float* __restrict__ selOb, const float* __restrict__ selBox,
      const int* __restrict__ selCls, const float* __restrict__ selScore,
      const float* __restrict__ U, float* __restrict__ outBox,
      float* __restrict__ outCls, float* __restrict__ outScore) {
    __shared__ float obx[PRE][4];
    __shared__ int   keep[PRE];
    __shared__ float sS[512];
    __shared__ int   sI[512];
    const int b = blockIdx.x;
    const int t = threadIdx.x;
    float bx0 = 0, bx1 = 0, bx2 = 0, bx3 = 0;
    if (t < PRE) {
        const float* o = selOb + ((size_t)b * PRE + t) * 4;
        bx0 = o[0]; bx1 = o[1]; bx2 = o[2]; bx3 = o[3];
        obx[t][0] = bx0; obx[t][1] = bx1; obx[t][2] = bx2; obx[t][3] = bx3;
        keep[t] = (selScore[b * PRE + t] > CONF) ? 1 : 0;
    }
    __syncthreads();
    const float* Ub = U + (size_t)b * PRE * PRE;
    for (int i = 0; i < PRE; ++i) {
        if (keep[i] && t > i && t < PRE && keep[t]) {
            float lx = fmaxf(bx0, obx[i][0]);
            float ly = fmaxf(bx1, obx[i][1]);
            float rx = fminf(bx2, obx[i][2]);
            float ry = fminf(bx3, obx[i][3]);
            float iw = fmaxf(rx - lx, 0.0f);
            float ih = fmaxf(ry - ly, 0.0f);
            float inter = iw * ih;
            float iou = inter / (Ub[(size_t)i * PRE + t] - inter + 1e-9f);
            if (iou > NMSTHR) keep[t] = 0;
        }
        __syncthreads();
    }
    // masked scores, pad 400..511 with -1, stable-desc bitonic sort
    sS[t] = (t < PRE) ? (keep[t] ? selScore[b * PRE + t] : 0.0f) : -1.0f;
    sI[t] = t;
    __syncthreads();
    for (int k = 2; k <= 512; k <<= 1) {
        for (int j = k >> 1; j > 0; j >>= 1) {
            int ixj = t ^ j;
            if (ixj > t) {
                float sa = sS[t], sb2 = sS[ixj];
                int   ia = sI[t], ib2 = sI[ixj];
                bool less = (sa < sb2) || (sa == sb2 && ia > ib2);
                bool up   = ((t & k) == 0);
                if (less == up) { sS[t] = sb2; sS[ixj] = sa; sI[t] = ib2; sI[ixj] = ia; }
            }
            __syncthreads();
        }
    }
    if (t < MAXD) {
        int j = sI[t];
        const float* cb = selBox + ((size_t)b * PRE + j) * 4;
        float* ob = outBox + ((size_t)b * MAXD + t) * 4;
        ob[0] = cb[0]; ob[1] = cb[1]; ob[2] = cb[2]; ob[3] = cb[3];
        outCls[b * MAXD + t]   = (float)selCls[b * PRE + j];
        outScore[b * MAXD + t] = sS[t];
    }
}

// ---------------- host side ----------------
extern "C" void kernel_launch(void* const* d_in, const int* in_sizes, int n_in,
                              void* d_out, int out_size, void* d_ws, size_t ws_size,
                              hipStream_t stream) {
    const float* logits = (const float*)d_in[0];
    const float* loc    = (const float*)d_in[1];
    const float* pri    = (const float*)d_in[2];
    const int P = in_sizes[2] / 4;

    char* w = (char*)d_ws;
    auto carve = [&](size_t bytes) -> char* {
        char* r = w;
        w += (bytes + 255) & ~(size_t)255;
        return r;
    };
    float*    dec      = (float*)   carve((size_t)B_ * P * 4 * 4);
    unsigned* hist     = (unsigned*)carve((size_t)B_ * NBINS * 4);
    int*      thr      = (int*)     carve((size_t)B_ * 4);
    int*      cnt      = (int*)     carve((size_t)B_ * 4);
    float*    candS    = (float*)   carve((size_t)B_ * CAP * 4);
    int*      candI    = (int*)     carve((size_t)B_ * CAP * 4);
    float*    selBox   = (float*)   carve((size_t)B_ * PRE * 4 * 4);
    float*    selOb    = (float*)   carve((size_t)B_ * PRE * 4 * 4);
    int*      selCls   = (int*)     carve((size_t)B_ * PRE * 4);
    float*    selScore = (float*)   carve((size_t)B_ * PRE * 4);
    float*    selArea  = (float*)   carve((size_t)B_ * PRE * 4);
    float*    U        = (float*)   carve((size_t)B_ * PRE * PRE * 4);

    k_init<<<(B_ * NBINS + 255) / 256, 256, 0, stream>>>(hist, cnt);
    k_decode<<<(B_ * P + 255) / 256, 256, 0, stream>>>(loc, pri, dec, P);
    dim3 g1((P + 63) / 64, B_);
    k_scores<0><<<g1, 256, 0, stream>>>(logits, hist, thr, cnt, candS, candI, P);
    k_thr<<<B_, 32, 0, stream>>>(hist, thr);
    k_scores<1><<<g1, 256, 0, stream>>>(logits, hist, thr, cnt, candS, candI, P);
    k_sel<<<B_, 1024, 0, stream>>>(cnt, candS, candI, dec, selBox, selOb, selCls, selScore, selArea, P);
    k_umat<<<dim3(PRE / 16, PRE / 16, B_), 32, 0, stream>>>(selArea, U);

    float* outBox   = (float*)d_out;
    float* outCls   = outBox + (size_t)B_ * MAXD * 4;
    float* outScore = outCls + (size_t)B_ * MAXD;
    k_nms<<<B_, 512, 0, stream>>>(selOb, selBox, selCls, selScore, U, outBox, outCls, outScore);
}